// GraphSAGE_46231027974474
// MI455X (gfx1250) — compile-verified
//
#include <hip/hip_runtime.h>
#include <hip/hip_bf16.h>

// GraphSAGE (2-layer, mean aggregation) for MI455X / gfx1250.
//
// Cost model: total working set (~112 MB) is resident in the 192 MB L2, so
// runtime is bound by the 153.6M global_atomic_add_f32 edge scatters, not the
// 3.3 GFLOP of dense GEMM. The GEMMs therefore use the *exact* fp32 WMMA path
// (V_WMMA_F32_16X16X4_F32): identical precision to the fp32 reference, and the
// matrix pipe is nowhere near the bottleneck so lower precision buys nothing.

typedef __attribute__((ext_vector_type(2))) float v2f;
typedef __attribute__((ext_vector_type(8))) float v8f;

// ---------------------------------------------------------------- zero fill
__global__ void zero_f32(float* __restrict__ p, long long n) {
  long long i = (long long)blockIdx.x * blockDim.x + threadIdx.x;
  long long stride = (long long)gridDim.x * blockDim.x;
  for (; i < n; i += stride) p[i] = 0.0f;
}

// ---------------------------------------------------------------- in-degree
__global__ void degree_f32(const long long* __restrict__ dst,
                           float* __restrict__ cnt, long long E) {
  long long i = (long long)blockIdx.x * blockDim.x + threadIdx.x;
  long long stride = (long long)gridDim.x * blockDim.x;
  for (; i < E; i += stride) {
    atomicAdd(&cnt[(int)dst[i]], 1.0f);
  }
}

// ------------------------------------------------- edge scatter: agg[dst] += feat[src]
// One wave32 per edge: src/dst index loads are wave-uniform (single merged
// fetch), each lane handles features {lane, lane+32} -> two coalesced 128B row
// reads and two coalesced global_atomic_add_f32 rows. All traffic L2-resident.
__global__ void scatter_add(const float* __restrict__ feat,
                            const long long* __restrict__ src,
                            const long long* __restrict__ dst,
                            float* __restrict__ agg, long long E) {
  const int lane = (int)(threadIdx.x & 31);
  long long e = (long long)blockIdx.x * (blockDim.x >> 5) + (threadIdx.x >> 5);
  const long long waves = (long long)gridDim.x * (blockDim.x >> 5);
  for (; e < E; e += waves) {
    const long long s = src[e];          // uniform across the wave
    const long long d = dst[e];          // uniform across the wave
    const float* __restrict__ xr = feat + (s << 6);
    float* __restrict__ ar = agg + (d << 6);
    const float v0 = xr[lane];
    const float v1 = xr[lane + 32];
    atomicAdd(&ar[lane], v0);
    atomicAdd(&ar[lane + 32], v1);
  }
}

// ------------------------------------------------- SAGE layer via fp32 WMMA
// One wave32 handles a tile of 16 nodes:
//   out[16 x DOUT] = (agg/deg)[16x64] @ Wl^T + x[16x64] @ Wr^T + bias
// K=64 split into 16 steps of V_WMMA_F32_16X16X4_F32; NT n-tiles of 16 output
// features (NT=4 -> 64, NT=3 -> 48 >= 40).
//
// Out-of-range output columns (NT*16 > DOUT) are handled WITHOUT divergence:
// the weight-row index is clamped (duplicate row computed, result discarded),
// and only the final store is lane-guarded. EXEC is all-ones for every WMMA.
//
// fp32 16x4 A layout (ISA 7.12.2): lane L holds row M = L&15;
//   lanes 0-15:  VGPR0 = K=k0,   VGPR1 = K=k0+1
//   lanes 16-31: VGPR0 = K=k0+2, VGPR1 = K=k0+3
// B (4x16) mirrors this with N = L&15. D: VGPR j -> (M = j + 8*(L>=16), N = L&15).
template <int NT, int DOUT, bool RELU>
__global__ void sage_wmma(const float* __restrict__ agg,
                          const float* __restrict__ cnt,
                          const float* __restrict__ xin,
                          const float* __restrict__ Wl,
                          const float* __restrict__ Wr,
                          const float* __restrict__ bias,
                          float* __restrict__ out,
                          int ntiles) {
  const int lane = (int)(threadIdx.x & 31);
  const int wid  = (int)(blockIdx.x * (blockDim.x >> 5) + (threadIdx.x >> 5));
  if (wid >= ntiles) return;          // whole-wave uniform exit: EXEC stays all-1s

  const int node0 = wid << 4;
  const int m  = lane & 15;           // A row / B column / D column index
  const int kh = (lane & 16) ? 2 : 0; // K-pair select per half-wave

  const int row = node0 + m;
  const float inv = 1.0f / fmaxf(cnt[row], 1.0f);   // mean; isolated nodes -> 0
  const float* __restrict__ arow = agg + ((long long)row << 6);
  const float* __restrict__ xrow = xin + ((long long)row << 6);

  // Weight-row bases: clamped, never divergent.
  const float* wlrow[NT];
  const float* wrrow[NT];
#pragma unroll
  for (int t = 0; t < NT; ++t) {
    int r = t * 16 + m;
    if (r > DOUT - 1) r = DOUT - 1;   // clamp: duplicate row, discarded at store
    wlrow[t] = Wl + (long long)r * 64;
    wrrow[t] = Wr + (long long)r * 64;
  }

  const v8f z8 = {};
  v8f acc[NT];
#pragma unroll
  for (int t = 0; t < NT; ++t) acc[t] = z8;

#pragma unroll
  for (int k0 = 0; k0 < 64; k0 += 4) {
    v2f aagg = *(const v2f*)(arow + k0 + kh);
    aagg = aagg * inv;                // fold mean-normalization into A rows
    const v2f ax = *(const v2f*)(xrow + k0 + kh);
#pragma unroll
    for (int t = 0; t < NT; ++t) {
      const v2f bl = *(const v2f*)(wlrow[t] + k0 + kh);
      const v2f br = *(const v2f*)(wrrow[t] + k0 + kh);
      acc[t] = __builtin_amdgcn_wmma_f32_16x16x4_f32(
          false, aagg, false, bl, (short)0, acc[t], false, false);
      acc[t] = __builtin_amdgcn_wmma_f32_16x16x4_f32(
          false, ax,   false, br, (short)0, acc[t], false, false);
    }
  }

  // D layout: VGPR j holds node (node0 + j + 8*(lane>=16)), feature t*16 + m.
  const int nodeBase = node0 + ((lane & 16) ? 8 : 0);
#pragma unroll
  for (int t = 0; t < NT; ++t) {
    const int f = t * 16 + m;
    const bool ok = (f < DOUT);       // compile-time true for full tiles
    const float bv = bias[ok ? f : (DOUT - 1)];
#pragma unroll
    for (int j = 0; j < 8; ++j) {
      float v = acc[t][j] + bv;
      if (RELU) v = fmaxf(v, 0.0f);
      if (ok) out[(long long)(nodeBase + j) * DOUT + f] = v;
    }
  }
}

// ------------------------------------------------------------------ launch
extern "C" void kernel_launch(void* const* d_in, const int* in_sizes, int n_in,
                              void* d_out, int out_size, void* d_ws, size_t ws_size,
                              hipStream_t stream) {
  const float*      x   = (const float*)d_in[0];
  const long long*  ei  = (const long long*)d_in[1];   // int64 [2, E]
  const float*      W1l = (const float*)d_in[2];
  const float*      W1r = (const float*)d_in[3];
  const float*      b1  = (const float*)d_in[4];
  const float*      W2l = (const float*)d_in[5];
  const float*      W2r = (const float*)d_in[6];
  const float*      b2  = (const float*)d_in[7];

  const long long E    = in_sizes[1] / 2;        // 1,200,000
  const int       N    = in_sizes[0] / 64;       // 100,000
  const long long Nf   = (long long)N * 64;

  const long long* src = ei;
  const long long* dst = ei + E;

  // Workspace layout: agg[N*64] | h[N*64] | cnt[N]   (~51.6 MB)
  char*  ws  = (char*)d_ws;
  float* agg = (float*)ws;
  float* h   = (float*)(ws + (size_t)Nf * sizeof(float));
  float* cnt = (float*)(ws + 2 * (size_t)Nf * sizeof(float));

  const int TPB = 256;
  const int tiles   = (N + 15) / 16;             // 6250 (exact: N % 16 == 0)
  const int sblocks = (tiles + 7) / 8;           // 8 waves per 256-thread block

  // ---- layer 1 ----
  zero_f32<<<2048, TPB, 0, stream>>>(agg, Nf);
  zero_f32<<<256,  TPB, 0, stream>>>(cnt, (long long)N);
  degree_f32<<<2048, TPB, 0, stream>>>(dst, cnt, E);
  scatter_add<<<8192, TPB, 0, stream>>>(x, src, dst, agg, E);
  sage_wmma<4, 64, true><<<sblocks, TPB, 0, stream>>>(agg, cnt, x, W1l, W1r,
                                                      b1, h, tiles);
  // ---- layer 2 ----
  zero_f32<<<2048, TPB, 0, stream>>>(agg, Nf);
  scatter_add<<<8192, TPB, 0, stream>>>(h, src, dst, agg, E);
  sage_wmma<3, 40, false><<<sblocks, TPB, 0, stream>>>(agg, cnt, h, W2l, W2r,
                                                       b2, (float*)d_out, tiles);
}